// Quantize_7842610283120
// MI455X (gfx1250) — compile-verified
//
#include <hip/hip_runtime.h>

// VQ-VAE quantize for MI455X (gfx1250, wave32).
// dist = ||f||^2 - 2 f.e + ||e||^2 ; ind = argmin ; q = embed[:,ind]
// diff = (1+BETA)*mean((q-x)^2) ; straight-through output == q.
//
// d_out layout (float):
//   [0, 4194304)   quantize  (16*32*32*256)
//   [4194304]      diff scalar
//   [4194305, +16384) embed_ind (as float)
//
// d_ws layout:
//   [0, 8 MB)            embedT (8192 x 256 fp32)
//   [8 MB, +32 KB)       enorm[8192]
//   [8 MB+32 KB, +64 KB) ind[16384] (int)

typedef float    v2f __attribute__((ext_vector_type(2)));
typedef float    v8f __attribute__((ext_vector_type(8)));
typedef int      v4i __attribute__((ext_vector_type(4)));
typedef unsigned v4u __attribute__((ext_vector_type(4)));

#define DIM   256
#define NEMB  8192
#define NROWS 16384
#define BETA  0.25f

#define PADF  4            // pad floats per LDS row (bank-conflict avoidance)
#define ROWF  (DIM + PADF) // 260 dwords per LDS row

// ---- CDNA5 async global->LDS copy (ASYNCcnt path), with safe fallback -----
__device__ __forceinline__ void async_cp16(void* lds, const void* gmem) {
#if __has_builtin(__builtin_amdgcn_global_load_async_to_lds_b128)
    __builtin_amdgcn_global_load_async_to_lds_b128(
        (__attribute__((address_space(1))) v4i*)(gmem),
        (__attribute__((address_space(3))) v4i*)(lds),
        /*imm offset=*/0, /*cpol=*/0);
#else
    *(v4u*)lds = *(const v4u*)gmem;            // synchronous fallback
#endif
}

__device__ __forceinline__ void wait_async0() {
#if __has_builtin(__builtin_amdgcn_global_load_async_to_lds_b128)
#if __has_builtin(__builtin_amdgcn_s_wait_asynccnt)
    __builtin_amdgcn_s_wait_asynccnt(0);
#else
    asm volatile("s_wait_asynccnt 0" ::: "memory");
#endif
#endif
}

// ---------------------------------------------------------------------------
// Kernel 1: transpose codebook, compute per-code squared norms, zero diff.
// ---------------------------------------------------------------------------
__global__ void vq_init_kernel(const float* __restrict__ embed,
                               float* __restrict__ embedT,
                               float* __restrict__ enorm,
                               float* __restrict__ diff_out) {
    const int j = blockIdx.x * blockDim.x + threadIdx.x;   // code 0..8191
    float nrm = 0.0f;
#pragma unroll 4
    for (int k = 0; k < DIM; ++k) {
        float v = embed[(size_t)k * NEMB + j];
        embedT[(size_t)j * DIM + k] = v;
        nrm += v * v;
    }
    enorm[j] = nrm;
    if (j == 0) *diff_out = 0.0f;
}

// ---------------------------------------------------------------------------
// Kernel 2: fused fp32-WMMA distance GEMM + argmin.
// Block = 128 threads (4 waves); wave w owns rows [blk*64 + w*16, +16).
// A (16x256 of x) register-resident (64 x v2f = 128 VGPRs), read once.
// B tile (16 codes x 256) async-DMA'd into double-buffered LDS and shared by
// all 4 waves; 64 x v_wmma_f32_16x16x4_f32 per 16-code chunk, with an 8-deep
// software pipeline of ds_load_b64 B fragments to keep DS latency off the
// matrix-pipe critical path.
// ---------------------------------------------------------------------------
__launch_bounds__(128)
__global__ void vq_argmin_wmma_kernel(const float* __restrict__ x,
                                      const float* __restrict__ embedT,
                                      const float* __restrict__ enorm,
                                      int*   __restrict__ ind_int,
                                      float* __restrict__ ind_f32) {
    __shared__ float btile[2][16 * ROWF];                  // 2 x 16.25 KB

    const int lane  = threadIdx.x & 31;
    const int wave  = threadIdx.x >> 5;
    const int rowbase = blockIdx.x * 64 + wave * 16;
    const int rlane = lane & 15;   // A row within tile; D column residue
    const int khalf = lane >> 4;   // K sub-pair select; D row half

    // ---- register-resident A operand ----
    const float* xrow = x + (size_t)(rowbase + rlane) * DIM + khalf * 2;
    v2f a[64];
#pragma unroll
    for (int kb = 0; kb < 64; ++kb)
        a[kb] = *(const v2f*)(xrow + kb * 4);

    float minD[8];
    int   minJ[8];
#pragma unroll
    for (int r = 0; r < 8; ++r) { minD[r] = 3.4e38f; minJ[r] = 0; }

    // cooperative async copy of one 16-code tile (16 KB) into LDS buffer
    auto copy_chunk = [&](int cb, int buf) {
        const char* gbase = (const char*)(embedT + (size_t)cb * DIM);
#pragma unroll
        for (int it = 0; it < 8; ++it) {
            const int i   = threadIdx.x + it * 128;   // 0..1023 b128 transfers
            const int row = i >> 6;                   // 0..15
            const int off = (i & 63) * 16;            // bytes within 1KB row
            async_cp16((char*)&btile[buf][row * ROWF] + off,
                       gbase + row * 1024 + off);
        }
    };

    // prologue: stage chunk 0
    copy_chunk(0, 0);
    wait_async0();
    __syncthreads();

    for (int cb16 = 0; cb16 < NEMB / 16; ++cb16) {
        // prefetch next chunk into the other buffer (overlaps the WMMA burst)
        if (cb16 + 1 < NEMB / 16)
            copy_chunk((cb16 + 1) * 16, (cb16 + 1) & 1);

        const int   cb  = cb16 * 16;
        const int   col = cb + rlane;                  // this lane's code col
        const float* bl = &btile[cb16 & 1][rlane * ROWF + khalf * 2];

        v8f acc = {0.f, 0.f, 0.f, 0.f, 0.f, 0.f, 0.f, 0.f};

        // 8-deep B-fragment pipeline: refill slot kb&7 with fragment kb+8
        // while the WMMA for kb issues -> DS loads stay >=4 instrs ahead.
        v2f bf[8];
#pragma unroll
        for (int kb = 0; kb < 8; ++kb)
            bf[kb] = *(const v2f*)(bl + kb * 4);       // ds_load_b64
#pragma unroll
        for (int kb = 0; kb < 64; ++kb) {
            v2f b = bf[kb & 7];
            if (kb + 8 < 64)
                bf[kb & 7] = *(const v2f*)(bl + (kb + 8) * 4);
            acc = __builtin_amdgcn_wmma_f32_16x16x4_f32(
                false, a[kb], false, b, (short)0, acc, false, false);
        }

        const float en = enorm[col];
#pragma unroll
        for (int r = 0; r < 8; ++r) {
            float d = en - 2.0f * acc[r];              // ||f||^2 dropped
            if (d < minD[r]) { minD[r] = d; minJ[r] = col; }
        }

        wait_async0();       // next tile landed in LDS
        __syncthreads();     // visible to all waves; old buffer reusable
    }

    // cross-lane argmin over the 16 lanes of each half (D layout: VGPR r,
    // half h -> row r + 8h, col lane&15); xor masks stay within halves.
#pragma unroll
    for (int m = 1; m < 16; m <<= 1) {
#pragma unroll
        for (int r = 0; r < 8; ++r) {
            float od = __shfl_xor(minD[r], m, 32);
            int   oj = __shfl_xor(minJ[r], m, 32);
            if (od < minD[r] || (od == minD[r] && oj < minJ[r])) {
                minD[r] = od;
                minJ[r] = oj;
            }
        }
    }

    if (rlane == 0) {
#pragma unroll
        for (int r = 0; r < 8; ++r) {
            const int row = rowbase + khalf * 8 + r;
            ind_int[row] = minJ[r];
            ind_f32[row] = (float)minJ[r];
        }
    }
}

// ---------------------------------------------------------------------------
// Kernel 3: gather winning codes, write quantize, accumulate scaled sq-error.
// ---------------------------------------------------------------------------
__global__ void vq_gather_diff_kernel(const float* __restrict__ x,
                                      const float* __restrict__ embedT,
                                      const int* __restrict__ ind,
                                      float* __restrict__ qout,
                                      float* __restrict__ diff_acc) {
    const int row = blockIdx.x;
    const int t   = threadIdx.x;          // 0..255 channel
    const int j   = ind[row];

    const float q  = embedT[(size_t)j * DIM + t];
    const float xv = x[(size_t)row * DIM + t];
    qout[(size_t)row * DIM + t] = q;

    float e = q - xv;
    float p = e * e;
#pragma unroll
    for (int m = 16; m >= 1; m >>= 1)
        p += __shfl_xor(p, m, 32);
    if ((t & 31) == 0)
        atomicAdd(diff_acc, p * ((1.0f + BETA) / (float)((size_t)NROWS * DIM)));
}

// ---------------------------------------------------------------------------
extern "C" void kernel_launch(void* const* d_in, const int* in_sizes, int n_in,
                              void* d_out, int out_size, void* d_ws, size_t ws_size,
                              hipStream_t stream) {
    const float* x     = (const float*)d_in[0];   // [16,32,32,256]
    const float* embed = (const float*)d_in[1];   // [256,8192]

    float* out      = (float*)d_out;
    float* qout     = out;                                  // 4194304 floats
    float* diff_out = out + (size_t)NROWS * DIM;            // 1 float
    float* ind_f32  = diff_out + 1;                         // 16384 floats

    char*  ws      = (char*)d_ws;
    float* embedT  = (float*)ws;                                        // 8 MB
    float* enorm   = (float*)(ws + (size_t)NEMB * DIM * sizeof(float)); // 32 KB
    int*   ind_int = (int*)(ws + (size_t)NEMB * DIM * sizeof(float)
                               + (size_t)NEMB * sizeof(float));         // 64 KB

    vq_init_kernel<<<NEMB / 256, 256, 0, stream>>>(embed, embedT, enorm, diff_out);
    vq_argmin_wmma_kernel<<<NROWS / 64, 128, 0, stream>>>(x, embedT, enorm,
                                                          ind_int, ind_f32);
    vq_gather_diff_kernel<<<NROWS, 256, 0, stream>>>(x, embedT, ind_int,
                                                     qout, diff_out);
}